// MyGaussianPDF_75479755260306
// MI455X (gfx1250) — compile-verified
//
#include <hip/hip_runtime.h>

// Streaming Gaussian PDF: out[i] = exp(-0.5 * 0.54 * ||x_i - mu||^2)
//                                = exp2(K * (dx*dx + dy*dy)),  K = -0.27 * log2(e)
//
// Memory-bound (192 MB traffic, ~0.11 GFLOP): optimize for HBM bandwidth.
//  - b128 loads/stores (4 points per thread: 2x float4 in, 1x float4 out)
//  - non-temporal cache policy (once-touched stream, footprint == full 192MB L2)
//  - v_exp_f32 hardware transcendental (one per point)
//  - wave32: 256-thread blocks = 8 waves

typedef float v4f __attribute__((ext_vector_type(4)));

#define GAUSS_K (-0.38952766f) /* -0.5f * 0.54f * 1.4426950408889634f */

__global__ __launch_bounds__(256) void gauss_pdf_v4_kernel(
    const v4f* __restrict__ x4,   // x viewed as float4: one float4 = 2 points
    const float* __restrict__ mu, // 2 floats, uniform -> scalar load
    v4f* __restrict__ out4,       // out viewed as float4: 4 results
    int n4)                       // number of 4-point groups
{
    int i = blockIdx.x * 256 + (int)threadIdx.x;
    if (i >= n4) return;

    const float mu0 = mu[0];
    const float mu1 = mu[1];

    // Two 128-bit NT loads: points 4i .. 4i+3
    v4f a = __builtin_nontemporal_load(&x4[2 * i]);
    v4f b = __builtin_nontemporal_load(&x4[2 * i + 1]);

    float dx0 = a.x - mu0, dy0 = a.y - mu1;
    float dx1 = a.z - mu0, dy1 = a.w - mu1;
    float dx2 = b.x - mu0, dy2 = b.y - mu1;
    float dx3 = b.z - mu0, dy3 = b.w - mu1;

    float s0 = __builtin_fmaf(dx0, dx0, dy0 * dy0);
    float s1 = __builtin_fmaf(dx1, dx1, dy1 * dy1);
    float s2 = __builtin_fmaf(dx2, dx2, dy2 * dy2);
    float s3 = __builtin_fmaf(dx3, dx3, dy3 * dy3);

    v4f r;
    r.x = __builtin_amdgcn_exp2f(GAUSS_K * s0);  // v_exp_f32
    r.y = __builtin_amdgcn_exp2f(GAUSS_K * s1);
    r.z = __builtin_amdgcn_exp2f(GAUSS_K * s2);
    r.w = __builtin_amdgcn_exp2f(GAUSS_K * s3);

    // 128-bit NT store
    __builtin_nontemporal_store(r, &out4[i]);
}

// Scalar tail for n % 4 != 0 (not hit for N = 2^24, kept for correctness).
__global__ __launch_bounds__(64) void gauss_pdf_tail_kernel(
    const float* __restrict__ x, const float* __restrict__ mu,
    float* __restrict__ out, int start, int n)
{
    int i = start + blockIdx.x * 64 + (int)threadIdx.x;
    if (i >= n) return;
    float dx = x[2 * i]     - mu[0];
    float dy = x[2 * i + 1] - mu[1];
    float s  = __builtin_fmaf(dx, dx, dy * dy);
    out[i]   = __builtin_amdgcn_exp2f(GAUSS_K * s);
}

extern "C" void kernel_launch(void* const* d_in, const int* in_sizes, int n_in,
                              void* d_out, int out_size, void* d_ws, size_t ws_size,
                              hipStream_t stream) {
    (void)n_in; (void)d_ws; (void)ws_size; (void)out_size;

    const float* x  = (const float*)d_in[0];   // [N, 2] float32
    const float* mu = (const float*)d_in[1];   // [2]    float32
    float* out      = (float*)d_out;           // [N]    float32

    const int n  = in_sizes[0] / 2;            // number of points
    const int n4 = n / 4;                      // 4-point groups
    const int rem = n - n4 * 4;

    if (n4 > 0) {
        const int blocks = (n4 + 255) / 256;
        gauss_pdf_v4_kernel<<<blocks, 256, 0, stream>>>(
            (const v4f*)x, mu, (v4f*)out, n4);
    }
    if (rem > 0) {
        gauss_pdf_tail_kernel<<<1, 64, 0, stream>>>(x, mu, out, n4 * 4, n);
    }
}